// BiGRU_7842610282761
// MI455X (gfx1250) — compile-verified
//
#include <hip/hip_runtime.h>
#include <math.h>

#define VOCAB 32000
#define HID 8
#define EMB 32
#define SEQ 64
#define BATCH 64
#define D 40            /* EMB + HID */
#define THDIM 16        /* 2 * HID   */
#define ROWS (SEQ*BATCH)

typedef float v2f __attribute__((ext_vector_type(2)));
typedef float v8f __attribute__((ext_vector_type(8)));

__device__ __forceinline__ float sigmoidf_fast(float x) {
    return 1.0f / (1.0f + __expf(-x));
}

// ---------------------------------------------------------------------------
// Phase 1: bidirectional GRU scan.
// 8 lanes cooperate on one (batch, direction) pair; lane owns hidden unit j.
// Weight columns fully unrolled into registers; candidate hidden values are
// exchanged with wave32 shuffles (no LDS, no barriers).
// Emits the PRE-update hidden state at each step (reference scan semantics).
// ---------------------------------------------------------------------------
__global__ __launch_bounds__(256)
void gru_scan_kernel(const int* __restrict__ x,
                     const float* __restrict__ emb,
                     const float* __restrict__ Wz1, const float* __restrict__ bz1,
                     const float* __restrict__ Wr1, const float* __restrict__ br1,
                     const float* __restrict__ Wh1, const float* __restrict__ bh1,
                     const float* __restrict__ Wz2, const float* __restrict__ bz2,
                     const float* __restrict__ Wr2, const float* __restrict__ br2,
                     const float* __restrict__ Wh2, const float* __restrict__ bh2,
                     float* __restrict__ total_h)
{
    const int tid   = blockIdx.x * blockDim.x + threadIdx.x;  // 0..1023
    const int group = tid >> 3;                               // 0..127
    const int j     = tid & 7;                                // owned hidden unit
    const int b     = group & 63;                             // batch element
    const int dir   = group >> 6;                             // 0=fwd, 1=bwd
    const int lbase = (threadIdx.x & 31) & ~7;                // group's first lane

    const float* Wz = dir ? Wz2 : Wz1;
    const float* Wr = dir ? Wr2 : Wr1;
    const float* Wh = dir ? Wh2 : Wh1;
    const float  bz = dir ? bz2[0] : bz1[0];
    const float  br = dir ? br2[0] : br1[0];
    const float  bh = dir ? bh2[j] : bh1[j];

    float wz[D], wr[D], wh[D];
#pragma unroll
    for (int i = 0; i < D; ++i) {
        wz[i] = Wz[i];
        wr[i] = Wr[i];
        wh[i] = Wh[i*HID + j];       // column j of [D, HID]
    }

    float h[HID];
#pragma unroll
    for (int i = 0; i < HID; ++i) h[i] = 0.0f;

    for (int t = 0; t < SEQ; ++t) {
        // forward consumes enc[t] and emits at t; backward consumes enc[T-1-t]
        // and (after the outer reverse) emits at the same index T-1-t.
        const int pos = dir ? (SEQ - 1 - t) : t;
        const int tok = x[pos*BATCH + b];
        const float4* ep = (const float4*)(emb + (long)tok * EMB);
        float e[EMB];
#pragma unroll
        for (int q = 0; q < EMB/4; ++q) {
            float4 v = ep[q];
            e[4*q+0] = v.x; e[4*q+1] = v.y; e[4*q+2] = v.z; e[4*q+3] = v.w;
        }

        float za = bz, ra = br;
#pragma unroll
        for (int i = 0; i < HID; ++i) {
            za = fmaf(h[i], wz[i], za);
            ra = fmaf(h[i], wr[i], ra);
        }
#pragma unroll
        for (int k = 0; k < EMB; ++k) {
            za = fmaf(e[k], wz[HID+k], za);
            ra = fmaf(e[k], wr[HID+k], ra);
        }
        const float z = sigmoidf_fast(za);
        const float r = sigmoidf_fast(ra);

        float ca = bh;
#pragma unroll
        for (int i = 0; i < HID; ++i) ca = fmaf(r * h[i], wh[i], ca);
#pragma unroll
        for (int k = 0; k < EMB; ++k) ca = fmaf(e[k], wh[HID+k], ca);
        const float hc = tanhf(ca);

        // emit PRE-update hidden state
        total_h[(pos*BATCH + b)*THDIM + dir*HID + j] = h[j];

        // h = (1-z)*h + z*h_cand  (gather the 8 candidates from the group)
#pragma unroll
        for (int i = 0; i < HID; ++i) {
            const float hci = __shfl(hc, lbase + i, 32);
            h[i] = fmaf(z, hci - h[i], h[i]);
        }
    }
}

// ---------------------------------------------------------------------------
// Phase 2: logits = total_h @ Wout + bout, then log_softmax over vocab.
// One block per 16-row tile; 8 waves stride the 2000 column tiles (250 each,
// exact division -> uniform EXEC around WMMA). K=16 handled as 4 chained
// V_WMMA_F32_16X16X4_F32 ops (pure f32 math). Two passes: online (m,s)
// reduction, then recompute + write logit - lse. Output write (524 MB) is the
// roofline; Wout (2 MB) stays L2-resident across all 256 blocks.
// ---------------------------------------------------------------------------
__global__ __launch_bounds__(256)
void logits_logsoftmax_kernel(const float* __restrict__ total_h,
                              const float* __restrict__ Wout,
                              const float* __restrict__ bout,
                              float* __restrict__ out)
{
    const int tileM  = blockIdx.x;          // 0..255
    const int lane   = threadIdx.x & 31;
    const int wave   = threadIdx.x >> 5;    // 0..7
    const int lo16   = lane & 15;
    const int hiHalf = lane >> 4;           // 0 or 1

    __shared__ float sh_m[8][16];
    __shared__ float sh_s[8][16];
    __shared__ float sh_lse[16];

    // A tile (16 rows x K=16 f32) as four K=4 chunks.
    // ISA layout: M = lane%16 for both halves; VGPR0 holds K=(lane<16?0:2),
    // VGPR1 holds K+1.
    const int rowA = tileM*16 + lo16;
    v2f a[4];
#pragma unroll
    for (int kk = 0; kk < 4; ++kk) {
        const int kb = 4*kk + 2*hiHalf;
        a[kk].x = total_h[rowA*THDIM + kb];
        a[kk].y = total_h[rowA*THDIM + kb + 1];
    }

    // per-lane online softmax state; C/D element i covers row (i + 8*hiHalf)
    float m[8], s[8];
#pragma unroll
    for (int i = 0; i < 8; ++i) { m[i] = -INFINITY; s[i] = 0.0f; }

    // ------------------- pass 1: running max / sum(exp) -------------------
    for (int ct = wave; ct < VOCAB/16; ct += 8) {
        const int col  = ct*16 + lo16;
        const float bb = bout[col];         // bias depends on column only
        v8f c;
#pragma unroll
        for (int i = 0; i < 8; ++i) c[i] = bb;
#pragma unroll
        for (int kk = 0; kk < 4; ++kk) {
            const int kb = 4*kk + 2*hiHalf;
            v2f bmat;
            bmat.x = Wout[kb*VOCAB + col];
            bmat.y = Wout[(kb+1)*VOCAB + col];
            c = __builtin_amdgcn_wmma_f32_16x16x4_f32(false, a[kk], false, bmat,
                                                      (short)0, c, false, false);
        }
#pragma unroll
        for (int i = 0; i < 8; ++i) {
            const float v  = c[i];
            const float nm = fmaxf(m[i], v);
            s[i] = s[i]*__expf(m[i]-nm) + __expf(v-nm);
            m[i] = nm;
        }
    }

    // reduce across the 16 lanes of each half-wave (same rows, disjoint cols)
#pragma unroll
    for (int mask = 1; mask < 16; mask <<= 1) {
#pragma unroll
        for (int i = 0; i < 8; ++i) {
            const float om = __shfl_xor(m[i], mask, 32);
            const float os = __shfl_xor(s[i], mask, 32);
            const float nm = fmaxf(m[i], om);
            s[i] = s[i]*__expf(m[i]-nm) + os*__expf(om-nm);
            m[i] = nm;
        }
    }
    if (lo16 == 0) {
#pragma unroll
        for (int i = 0; i < 8; ++i) {
            sh_m[wave][hiHalf*8 + i] = m[i];
            sh_s[wave][hiHalf*8 + i] = s[i];
        }
    }
    __syncthreads();
    if (threadIdx.x < 16) {
        float M = -INFINITY;
#pragma unroll
        for (int w = 0; w < 8; ++w) M = fmaxf(M, sh_m[w][threadIdx.x]);
        float S = 0.0f;
#pragma unroll
        for (int w = 0; w < 8; ++w) S += sh_s[w][threadIdx.x] * __expf(sh_m[w][threadIdx.x] - M);
        sh_lse[threadIdx.x] = M + __logf(S);
    }
    __syncthreads();

    float lse[8];
#pragma unroll
    for (int i = 0; i < 8; ++i) lse[i] = sh_lse[hiHalf*8 + i];

    // --------------- pass 2: recompute logits, write log-probs ---------------
    const long rowBase = (long)(tileM*16 + hiHalf*8);
    for (int ct = wave; ct < VOCAB/16; ct += 8) {
        const int col  = ct*16 + lo16;
        const float bb = bout[col];
        v8f c;
#pragma unroll
        for (int i = 0; i < 8; ++i) c[i] = bb;
#pragma unroll
        for (int kk = 0; kk < 4; ++kk) {
            const int kb = 4*kk + 2*hiHalf;
            v2f bmat;
            bmat.x = Wout[kb*VOCAB + col];
            bmat.y = Wout[(kb+1)*VOCAB + col];
            c = __builtin_amdgcn_wmma_f32_16x16x4_f32(false, a[kk], false, bmat,
                                                      (short)0, c, false, false);
        }
#pragma unroll
        for (int i = 0; i < 8; ++i) {
            out[(rowBase + i)*VOCAB + col] = c[i] - lse[i];
        }
    }
}

// ---------------------------------------------------------------------------
extern "C" void kernel_launch(void* const* d_in, const int* in_sizes, int n_in,
                              void* d_out, int out_size, void* d_ws, size_t ws_size,
                              hipStream_t stream)
{
    const int*   x    = (const int*)  d_in[0];
    const float* emb  = (const float*)d_in[1];
    const float* Wz1  = (const float*)d_in[2];
    const float* bz1  = (const float*)d_in[3];
    const float* Wr1  = (const float*)d_in[4];
    const float* br1  = (const float*)d_in[5];
    const float* Wh1  = (const float*)d_in[6];
    const float* bh1  = (const float*)d_in[7];
    const float* Wz2  = (const float*)d_in[8];
    const float* bz2  = (const float*)d_in[9];
    const float* Wr2  = (const float*)d_in[10];
    const float* br2  = (const float*)d_in[11];
    const float* Wh2  = (const float*)d_in[12];
    const float* bh2  = (const float*)d_in[13];
    const float* Wout = (const float*)d_in[14];
    const float* bout = (const float*)d_in[15];
    float* out = (float*)d_out;

    float* total_h = (float*)d_ws;   // [SEQ][BATCH][16] f32 = 256 KB scratch

    gru_scan_kernel<<<4, 256, 0, stream>>>(x, emb,
                                           Wz1, bz1, Wr1, br1, Wh1, bh1,
                                           Wz2, bz2, Wr2, br2, Wh2, bh2,
                                           total_h);

    logits_logsoftmax_kernel<<<ROWS/16, 256, 0, stream>>>(total_h, Wout, bout, out);
}